// KalmanFilter_35665408426063
// MI455X (gfx1250) — compile-verified
//
#include <hip/hip_runtime.h>
#include <math.h>

// Problem constants (match reference)
#define BSZ  256
#define TN   100
#define DZv  32
#define DAv  16
#define DUv  2
#define KN   8
#define DOBSv 8
#define HN   64
#define DXv  (DAv + DOBSv)   // 24

typedef __attribute__((ext_vector_type(2))) float v2f;
typedef __attribute__((ext_vector_type(8))) float v8f;

// ---------------------------------------------------------------------------
// Tiled f32 WMMA matmul on LDS-resident matrices.
//   D[M x N] = A[M x Kd] * B (+ D if ACC)
// B is either row-major [Kd x N] (TRB=false) or sourced transposed from a
// row-major [N x Kd] buffer (TRB=true).
// Fragment layouts per CDNA5 ISA 7.12.2 (wave32, V_WMMA_F32_16X16X4_F32):
//   A 16x4 : lanes 0-15 -> M=lane, VGPR0/1 = K{0,1}; lanes 16-31 -> K{2,3}
//   B 4x16 : lanes 0-15 -> N=lane, VGPR0/1 = K{0,1}; lanes 16-31 -> K{2,3}
//   C/D    : lanes 0-15 -> N=lane, VGPR r = M=r;  lanes 16-31 -> M=8+r
// All call sites are in uniform control flow => EXEC all ones at the WMMA.
// ---------------------------------------------------------------------------
template<int M, int N, int Kd, bool TRB, bool ACC>
__device__ __forceinline__ void mm_wmma(const float* __restrict__ A, int lda,
                                        const float* __restrict__ Bm, int ldb,
                                        float* __restrict__ D, int ldd, int lane)
{
    const int  col = lane & 15;
    const int  kh  = lane >> 4;          // which K-half this lane supplies
    const bool lo  = (lane < 16);
#pragma unroll
    for (int mo = 0; mo < M; mo += 16) {
#pragma unroll
        for (int no = 0; no < N; no += 16) {
            v8f c;
#pragma unroll
            for (int r = 0; r < 8; ++r) {
                int rr = lo ? r : (r + 8);
                c[r] = ACC ? D[(mo + rr) * ldd + no + col] : 0.0f;
            }
#pragma unroll
            for (int ko = 0; ko < Kd; ko += 4) {
                v2f a, bf;
                a[0] = A[(mo + col) * lda + (ko + 2 * kh + 0)];
                a[1] = A[(mo + col) * lda + (ko + 2 * kh + 1)];
                if (TRB) {
                    bf[0] = Bm[(no + col) * ldb + (ko + 2 * kh + 0)];
                    bf[1] = Bm[(no + col) * ldb + (ko + 2 * kh + 1)];
                } else {
                    bf[0] = Bm[(ko + 2 * kh + 0) * ldb + (no + col)];
                    bf[1] = Bm[(ko + 2 * kh + 1) * ldb + (no + col)];
                }
                c = __builtin_amdgcn_wmma_f32_16x16x4_f32(
                        /*neg_a=*/false, a, /*neg_b=*/false, bf,
                        /*c_mod=*/(short)0, c, /*reuse_a=*/false, /*reuse_b=*/false);
            }
#pragma unroll
            for (int r = 0; r < 8; ++r) {
                int rr = lo ? r : (r + 8);
                D[(mo + rr) * ldd + no + col] = c[r];
            }
        }
    }
    __syncthreads();
}

__device__ __forceinline__ float sigmoidf_(float x) { return 1.0f / (1.0f + __expf(-x)); }

// GRU step + softmax gate: consumes sx[24], sh[64]; updates sh (h2) and
// salpha[8] (softmax logits -> probabilities). Single wave32.
__device__ __forceinline__ void gru_alpha(
    int lane,
    const float* __restrict__ Wx, const float* __restrict__ Wh,
    const float* __restrict__ bx, const float* __restrict__ bh,
    const float* __restrict__ oW, const float* __restrict__ ob,
    float* __restrict__ sx, float* __restrict__ sh,
    float* __restrict__ sgx, float* __restrict__ sgh,
    float* __restrict__ salpha)
{
    // gx = Wx @ x + bx, gh = Wh @ h + bh   (192 rows, 6 per lane)
#pragma unroll
    for (int j = lane; j < 3 * HN; j += 32) {
        float ax = bx[j];
#pragma unroll
        for (int i = 0; i < DXv; ++i) ax += Wx[j * DXv + i] * sx[i];
        sgx[j] = ax;
        float ah = bh[j];
#pragma unroll
        for (int i = 0; i < HN; ++i) ah += Wh[j * HN + i] * sh[i];
        sgh[j] = ah;
    }
    __syncthreads();
    // gates (2 hidden elements per lane)
    float hnew[2];
#pragma unroll
    for (int e = 0; e < 2; ++e) {
        int idx = lane + 32 * e;
        float rg = sigmoidf_(sgx[idx] + sgh[idx]);
        float ug = sigmoidf_(sgx[HN + idx] + sgh[HN + idx]);
        float ng = tanhf(sgx[2 * HN + idx] + rg * sgh[2 * HN + idx]);
        hnew[e] = (1.0f - ug) * ng + ug * sh[idx];
    }
    __syncthreads();
    sh[lane] = hnew[0];
    sh[lane + 32] = hnew[1];
    __syncthreads();
    // logits
    if (lane < KN) {
        float lg = ob[lane];
#pragma unroll
        for (int j = 0; j < HN; ++j) lg += sh[j] * oW[j * KN + lane];
        salpha[lane] = lg;
    }
    __syncthreads();
    // softmax (redundant per-lane over K=8)
    float mx = salpha[0];
#pragma unroll
    for (int k = 1; k < KN; ++k) mx = fmaxf(mx, salpha[k]);
    float ex = 0.0f, sum = 0.0f;
#pragma unroll
    for (int k = 0; k < KN; ++k) {
        float e = __expf(salpha[k] - mx);
        sum += e;
        if (k == (lane & 7)) ex = e;
    }
    __syncthreads();
    if (lane < KN) salpha[lane] = ex / sum;
    __syncthreads();
}

__global__ __launch_bounds__(32) void kf_scan_kernel(
    const float* __restrict__ a_seq,  const float* __restrict__ a_var,
    const float* __restrict__ h_obs,  const float* __restrict__ Amat,
    const float* __restrict__ Cmat,   const float* __restrict__ Bmat,
    const float* __restrict__ u_seq,  const float* __restrict__ maskp,
    const float* __restrict__ z0p,    const float* __restrict__ P0ld,
    const float* __restrict__ a0p,    const float* __restrict__ logQ,
    const float* __restrict__ Wx,     const float* __restrict__ Wh,
    const float* __restrict__ bx,     const float* __restrict__ bh,
    const float* __restrict__ oW,     const float* __restrict__ ob,
    float* __restrict__ out)
{
    const int b    = blockIdx.x;
    const int lane = threadIdx.x;

    __shared__ float sP  [DZv * DZv];   // carry covariance P (then P_p)
    __shared__ float sPf [DZv * DZv];   // filtered covariance
    __shared__ float sT1 [DZv * DZv];   // temp product
    __shared__ float sIKC[DZv * DZv];   // I - Kg C
    __shared__ float sA2 [DZv * DZv];   // alpha2-mixed A
    __shared__ float sCm [DAv * DZv];   // mixed C (16x32)
    __shared__ float sPCt[DZv * DAv];   // P C^T (32x16), later reused as Kg*R
    __shared__ float sKg [DZv * DAv];   // Kalman gain (32x16)
    __shared__ float sSg [DAv * DZv];   // augmented [S | I] (16x32)
    __shared__ float sBm [DZv * DUv];   // mixed B (32x2)
    __shared__ float sz  [DZv];
    __shared__ float szf [DZv];
    __shared__ float szp [DZv];
    __shared__ float sh  [HN];
    __shared__ float sx  [DXv];
    __shared__ float sgx [3 * HN];
    __shared__ float sgh [3 * HN];
    __shared__ float salpha[KN];
    __shared__ float sahat[DAv], srv[DAv], saf[DAv], sap[DAv], sRd[DAv];

    // output section offsets (flat tuple: z_f, P_f, z_p, P_p, a_f, a_p, alpha2)
    const size_t BT   = (size_t)BSZ * TN;
    const size_t o_zf = 0;
    const size_t o_Pf = o_zf + BT * DZv;
    const size_t o_zp = o_Pf + BT * DZv * DZv;
    const size_t o_Pp = o_zp + BT * DZv;
    const size_t o_af = o_Pp + BT * DZv * DZv;
    const size_t o_ap = o_af + BT * DAv;
    const size_t o_al = o_ap + BT * DAv;

    // ---- initial carry: alpha0/h0 from GRU(a_0, h_obs), z0, P0 = diag(exp) ----
    sh[lane] = 0.0f;  sh[lane + 32] = 0.0f;
    if (lane < DAv)            sx[lane] = a0p[lane];
    else if (lane < DXv)       sx[lane] = h_obs[b * DOBSv + (lane - DAv)];
    __syncthreads();
    gru_alpha(lane, Wx, Wh, bx, bh, oW, ob, sx, sh, sgx, sgh, salpha);

    sz[lane] = z0p[lane];
    for (int e = lane; e < DZv * DZv; e += 32) {
        int i = e >> 5, j = e & 31;
        sP[e] = (i == j) ? __expf(P0ld[i]) : 0.0f;
    }
    __syncthreads();

    for (int t = 0; t < TN; ++t) {
        const float* a_k = a_seq + ((size_t)b * TN + t) * DAv;
        const float* Rdg = a_var + ((size_t)b * TN + t) * DAv;
        const float* u_k = u_seq + ((size_t)b * TN + t) * DUv;
        const float  mval = maskp[(size_t)b * TN + t];

        // C = sum_k alpha[k] * C_k  (16x32)
        for (int e = lane; e < DAv * DZv; e += 32) {
            float acc = 0.0f;
#pragma unroll
            for (int k = 0; k < KN; ++k) acc += salpha[k] * Cmat[k * DAv * DZv + e];
            sCm[e] = acc;
        }
        __syncthreads();

        // a_hat = C z ; residual ; R diag
        if (lane < DAv) {
            float acc = 0.0f;
#pragma unroll
            for (int j = 0; j < DZv; ++j) acc += sCm[lane * DZv + j] * sz[j];
            sahat[lane] = acc;
            srv[lane]   = a_k[lane] - acc;
            sRd[lane]   = Rdg[lane];
        }
        __syncthreads();

        // PCt = P @ C^T            [32x16]
        mm_wmma<DZv, DAv, DZv, true,  false>(sP,  DZv, sCm, DZv, sPCt, DAv, lane);
        // S = C @ PCt              [16x16] (into left half of augmented)
        mm_wmma<DAv, DAv, DZv, false, false>(sCm, DZv, sPCt, DAv, sSg, DZv, lane);
        // S += diag(R); right half := I
        for (int e = lane; e < DAv * DAv; e += 32) {
            int i = e >> 4, j = e & 15;
            if (i == j) sSg[i * DZv + j] += sRd[i];
            sSg[i * DZv + DAv + j] = (i == j) ? 1.0f : 0.0f;
        }
        __syncthreads();

        // Gauss-Jordan inverse of SPD S (lanes = 32 augmented columns)
        for (int piv = 0; piv < DAv; ++piv) {
            float pinv = 1.0f / sSg[piv * DZv + piv];
            float rowv = sSg[piv * DZv + lane] * pinv;
            __syncthreads();
            sSg[piv * DZv + lane] = rowv;
            __syncthreads();
#pragma unroll
            for (int r2 = 0; r2 < DAv; ++r2) {
                if (r2 == piv) continue;
                float f = sSg[r2 * DZv + piv];      // wave-uniform read
                sSg[r2 * DZv + lane] -= f * rowv;   // lockstep within wave
            }
            __syncthreads();
        }

        // Kg = PCt @ S^{-1}  (S symmetric => matches solve(S^T, C P^T)^T), then mask
        mm_wmma<DZv, DAv, DAv, false, false>(sPCt, DAv, sSg + DAv, DZv, sKg, DAv, lane);
        for (int e = lane; e < DZv * DAv; e += 32) sKg[e] *= mval;
        __syncthreads();

        // z_f = z + Kg r
        {
            float acc = sz[lane];
#pragma unroll
            for (int j = 0; j < DAv; ++j) acc += sKg[lane * DAv + j] * srv[j];
            szf[lane] = acc;
        }
        __syncthreads();

        // IKC = I - Kg @ C
        mm_wmma<DZv, DZv, DAv, false, false>(sKg, DAv, sCm, DZv, sT1, DZv, lane);
        for (int e = lane; e < DZv * DZv; e += 32) {
            int i = e >> 5, j = e & 31;
            sIKC[e] = ((i == j) ? 1.0f : 0.0f) - sT1[e];
        }
        __syncthreads();

        // Joseph form: Pf = IKC P IKC^T + (Kg R) Kg^T
        mm_wmma<DZv, DZv, DZv, false, false>(sIKC, DZv, sP,   DZv, sT1, DZv, lane);
        mm_wmma<DZv, DZv, DZv, true,  false>(sT1,  DZv, sIKC, DZv, sPf, DZv, lane);
        for (int e = lane; e < DZv * DAv; e += 32) sPCt[e] = sKg[e] * sRd[e & 15];
        __syncthreads();
        mm_wmma<DZv, DZv, DAv, true,  true >(sPCt, DAv, sKg,  DAv, sPf, DZv, lane);

        // a_f = C z_f ; a_for_alpha = m a_k + (1-m) a_hat
        if (lane < DAv) {
            float acc = 0.0f;
#pragma unroll
            for (int j = 0; j < DZv; ++j) acc += sCm[lane * DZv + j] * szf[j];
            saf[lane] = acc;
            sx[lane]  = mval * a_k[lane] + (1.0f - mval) * sahat[lane];
        } else if (lane < DXv) {
            sx[lane] = h_obs[b * DOBSv + (lane - DAv)];
        }
        __syncthreads();

        // alpha2, h2
        gru_alpha(lane, Wx, Wh, bx, bh, oW, ob, sx, sh, sgx, sgh, salpha);

        // mix A2, C2, B2 with alpha2
        for (int e = lane; e < DZv * DZv; e += 32) {
            float acc = 0.0f;
#pragma unroll
            for (int k = 0; k < KN; ++k) acc += salpha[k] * Amat[k * DZv * DZv + e];
            sA2[e] = acc;
        }
        for (int e = lane; e < DAv * DZv; e += 32) {
            float acc = 0.0f;
#pragma unroll
            for (int k = 0; k < KN; ++k) acc += salpha[k] * Cmat[k * DAv * DZv + e];
            sCm[e] = acc;
        }
        for (int e = lane; e < DZv * DUv; e += 32) {
            float acc = 0.0f;
#pragma unroll
            for (int k = 0; k < KN; ++k) acc += salpha[k] * Bmat[k * DZv * DUv + e];
            sBm[e] = acc;
        }
        __syncthreads();

        // z_p = A2 z_f + B2 u
        {
            float acc = 0.0f;
#pragma unroll
            for (int j = 0; j < DZv; ++j) acc += sA2[lane * DZv + j] * szf[j];
#pragma unroll
            for (int j = 0; j < DUv; ++j) acc += sBm[lane * DUv + j] * u_k[j];
            szp[lane] = acc;
        }
        __syncthreads();

        // P_p = A2 Pf A2^T + diag(exp(logQ))   (into carry sP)
        mm_wmma<DZv, DZv, DZv, false, false>(sA2, DZv, sPf, DZv, sT1, DZv, lane);
        mm_wmma<DZv, DZv, DZv, true,  false>(sT1, DZv, sA2, DZv, sP,  DZv, lane);
        sP[lane * DZv + lane] += __expf(logQ[lane]);   // 32 lanes == 32 diag elems
        // a_p = C2 z_p
        if (lane < DAv) {
            float acc = 0.0f;
#pragma unroll
            for (int j = 0; j < DZv; ++j) acc += sCm[lane * DZv + j] * szp[j];
            sap[lane] = acc;
        }
        __syncthreads();

        // ---- write step outputs ----
        const size_t bt = (size_t)b * TN + t;
        out[o_zf + bt * DZv + lane] = szf[lane];
        out[o_zp + bt * DZv + lane] = szp[lane];
        for (int e = lane; e < DZv * DZv; e += 32) {
            out[o_Pf + bt * (DZv * DZv) + e] = sPf[e];
            out[o_Pp + bt * (DZv * DZv) + e] = sP[e];
        }
        if (lane < DAv) {
            out[o_af + bt * DAv + lane] = saf[lane];
            out[o_ap + bt * DAv + lane] = sap[lane];
        }
        if (lane < KN) out[o_al + bt * KN + lane] = salpha[lane];

        // carry z
        sz[lane] = szp[lane];
        __syncthreads();
    }
}

extern "C" void kernel_launch(void* const* d_in, const int* in_sizes, int n_in,
                              void* d_out, int out_size, void* d_ws, size_t ws_size,
                              hipStream_t stream) {
    (void)in_sizes; (void)n_in; (void)out_size; (void)d_ws; (void)ws_size;
    const float* a_seq = (const float*)d_in[0];
    const float* a_var = (const float*)d_in[1];
    const float* h_obs = (const float*)d_in[2];
    const float* Amat  = (const float*)d_in[3];
    const float* Cmat  = (const float*)d_in[4];
    const float* Bmat  = (const float*)d_in[5];
    const float* u_seq = (const float*)d_in[6];
    const float* maskp = (const float*)d_in[7];
    const float* z0p   = (const float*)d_in[8];
    const float* P0ld  = (const float*)d_in[9];
    const float* a0p   = (const float*)d_in[10];
    const float* logQ  = (const float*)d_in[11];
    const float* Wx    = (const float*)d_in[12];
    const float* Wh    = (const float*)d_in[13];
    const float* bx    = (const float*)d_in[14];
    const float* bh    = (const float*)d_in[15];
    const float* oW    = (const float*)d_in[16];
    const float* ob    = (const float*)d_in[17];
    float* out = (float*)d_out;

    dim3 grid(BSZ), block(32);
    hipLaunchKernelGGL(kf_scan_kernel, grid, block, 0, stream,
                       a_seq, a_var, h_obs, Amat, Cmat, Bmat, u_seq, maskp,
                       z0p, P0ld, a0p, logQ, Wx, Wh, bx, bh, oW, ob, out);
}